// Network_31945966748134
// MI455X (gfx1250) — compile-verified
//
#include <hip/hip_runtime.h>

// MI455X / gfx1250, wave32.
// Roofline: 8.39M rows * (16B in + 12B out) = 235 MB @ 23.3 TB/s => ~10us.
// Bandwidth-bound streaming kernel; the [B,4]x[4,1] input GEMV is done with
// V_WMMA_F32_16X16X4_F32 (2 WMMAs per wave = 32 rows), everything else is
// cheap VALU. A-tile loads are b64/lane, 256B contiguous per wave per tile.
// All weights (w_in, w_hidden, w_out) are uniform -> scalar (s_load) path.

typedef __attribute__((ext_vector_type(2))) float v2f;
typedef __attribute__((ext_vector_type(4))) float v4f;
typedef __attribute__((ext_vector_type(8))) float v8f;

#define N_HID 9

__device__ __forceinline__ void epilogue(float dot,
                                         const float* __restrict__ wh,
                                         const float* __restrict__ wo,
                                         float* __restrict__ o) {
  // relu chain in reference order (exact rounding match where not zeroed)
  float h = fmaxf(dot, 0.0f);
#pragma unroll
  for (int i = 0; i < N_HID; ++i) h = fmaxf(h * wh[i], 0.0f);
  // out layer + softmax(dim=1) with max subtraction (as jax.nn.softmax)
  float l0 = h * wo[0];
  float l1 = h * wo[1];
  float l2 = h * wo[2];
  float mx = fmaxf(l0, fmaxf(l1, l2));
  float e0 = __expf(l0 - mx);
  float e1 = __expf(l1 - mx);
  float e2 = __expf(l2 - mx);
  float inv = 1.0f / (e0 + e1 + e2);
  o[0] = e0 * inv;
  o[1] = e1 * inv;
  o[2] = e2 * inv;
}

__global__ __launch_bounds__(256) void mlp_softmax_kernel(
    const float* __restrict__ x, const float* __restrict__ w_in,
    const float* __restrict__ w_hidden, const float* __restrict__ w_out,
    float* __restrict__ out, int nrows) {
  const int lane = threadIdx.x & 31;           // wave32
  const int wave = threadIdx.x >> 5;           // 8 waves / block
  const long long rowBase = ((long long)blockIdx.x * 8 + wave) * 32;

  // Uniform weight fetch: compiles to s_load_b128 (KM path), not vector loads.
  const float w0 = w_in[0];
  const float w1 = w_in[1];
  const float w2 = w_in[2];
  const float w3 = w_in[3];

  if (rowBase + 32 <= (long long)nrows) {
    // ---------- WMMA path: 32 rows per wave via two 16x16x4 f32 WMMAs ------
    const bool hi = (lane >= 16);

    // B (4x16): every column n holds w_in[k]; lane supplies its K-pair.
    // Value select on SGPRs -> v_cndmask, no divergent loads.
    v2f bmat;
    bmat.x = hi ? w2 : w0;
    bmat.y = hi ? w3 : w1;

    // A (16x4): lane L -> row (L&15), K pair {0,1} (lanes 0-15) / {2,3} (16-31)
    const int m = lane & 15;
    const float* p1 = x + (rowBase + m) * 4 + (hi ? 2 : 0);
    v2f a1 = *(const v2f*)p1;          // tile rows rowBase .. rowBase+15
    v2f a2 = *(const v2f*)(p1 + 64);   // tile rows rowBase+16 .. rowBase+31

    v8f cz = {};
    v8f d1 = __builtin_amdgcn_wmma_f32_16x16x4_f32(
        false, a1, false, bmat, (short)0, cz, false, false);
    v8f d2 = __builtin_amdgcn_wmma_f32_16x16x4_f32(
        false, a2, false, bmat, (short)0, cz, false, false);

    // Bijective lane -> row mapping so each of the 32 rows is finished once:
    //   lanes 0-7   : d1[j]  -> rows 0-7
    //   lanes 8-15  : d2[j]  -> rows 16-23
    //   lanes 16-23 : d1[j]  -> rows 8-15
    //   lanes 24-31 : d2[j]  -> rows 24-31
    const int j = lane & 7;
    v8f ds = ((lane >> 3) & 1) ? d2 : d1;
    float dot = ds[0];
#pragma unroll
    for (int k = 1; k < 8; ++k) dot = (j == k) ? ds[k] : dot;

    const int rofs = (lane & 7) + ((lane & 16) ? 8 : 0) + ((lane & 8) ? 16 : 0);
    epilogue(dot, w_hidden, w_out, out + (rowBase + rofs) * 3);
  } else {
    // ---------- scalar tail (wave-uniform branch; keeps WMMA EXEC all-1s) --
    long long row = rowBase + lane;
    if (row < (long long)nrows) {
      v4f v = *(const v4f*)(x + row * 4);
      float dot = v.x * w0 + v.y * w1 + v.z * w2 + v.w * w3;
      epilogue(dot, w_hidden, w_out, out + row * 3);
    }
  }
}

extern "C" void kernel_launch(void* const* d_in, const int* in_sizes, int n_in,
                              void* d_out, int out_size, void* d_ws,
                              size_t ws_size, hipStream_t stream) {
  const float* x        = (const float*)d_in[0];  // [B,4]
  const float* w_in     = (const float*)d_in[1];  // [1,4]
  const float* w_hidden = (const float*)d_in[2];  // [9]
  const float* w_out    = (const float*)d_in[3];  // [3,1]
  float* out            = (float*)d_out;          // [B,3]

  const int nrows  = in_sizes[0] / 4;             // B
  const int blocks = (nrows + 255) / 256;         // 256 rows per block
  mlp_softmax_kernel<<<blocks, 256, 0, stream>>>(x, w_in, w_hidden, w_out, out,
                                                 nrows);
}